// NearestNeighborAttention_75737453298160
// MI455X (gfx1250) — compile-verified
//
#include <hip/hip_runtime.h>

#define GRID_N 20
#define S_PTS (GRID_N * GRID_N * GRID_N)   // 8000
#define FDIM 512
#define NHEADS 8
#define HD 64
#define KNBR 32
#define BATCH 4

typedef __attribute__((ext_vector_type(2))) float v2f;
typedef __attribute__((ext_vector_type(8))) float v8f;

// ---------------------------------------------------------------------------
// Kernel 1: fused Q/K/V projection, fp32 WMMA (V_WMMA_F32_16X16X4_F32).
// One wave computes a 64x64 tile (4x4 WMMA register blocking) of
//   out = x[b] (S x F) * W^T (F x N)
// => 0.5 VMEM loads per WMMA; each A fragment reused across 4 column tiles,
//    each B fragment across 4 row tiles; L2 traffic halved vs 2x2 blocking.
// A 64-wide N tile is exactly one head, so the head-major scatter is trivial.
// blockIdx.x = S-tile/64 (125), blockIdx.y = head (8), blockIdx.z = b*3+which.
// ---------------------------------------------------------------------------
__global__ __launch_bounds__(32)
void qkv_proj_wmma(const float* __restrict__ x,
                   const float* __restrict__ Wq,
                   const float* __restrict__ Wk,
                   const float* __restrict__ Wv,
                   float* __restrict__ qb,
                   float* __restrict__ kb,
                   float* __restrict__ vb)
{
    const int lane  = threadIdx.x;          // 0..31 (wave32)
    const int sBase = blockIdx.x * 64;
    const int h     = blockIdx.y;           // nBase = h*64
    const int bz    = blockIdx.z;
    const int b     = bz / 3;
    const int which = bz - b * 3;

    const float* __restrict__ W   = (which == 0) ? Wq : (which == 1) ? Wk : Wv;
    float*       __restrict__ dst = (which == 0) ? qb : (which == 1) ? kb : vb;

    // A fragment: lane L -> row (L&15), K pair offset (L>>4)*2  (contiguous float2)
    // B fragment: lane L -> col (L&15), same K pair             (contiguous float2 of W row)
    const int row  = lane & 15;
    const int koff = (lane >> 4) << 1;      // 0 or 2

    const float* __restrict__ aptr = x + ((size_t)b * S_PTS + (sBase + row)) * FDIM + koff;
    const float* __restrict__ bptr = W + (size_t)(h * HD + row) * FDIM + koff;

    v8f acc[4][4];
#pragma unroll
    for (int r = 0; r < 4; ++r)
#pragma unroll
        for (int c = 0; c < 4; ++c)
            acc[r][c] = (v8f){};

#pragma unroll 2
    for (int kk = 0; kk < FDIM; kk += 4) {
        v2f af[4], bf[4];
#pragma unroll
        for (int r = 0; r < 4; ++r)
            af[r] = *(const v2f*)(aptr + (size_t)r * 16 * FDIM + kk);   // global_load_b64
#pragma unroll
        for (int c = 0; c < 4; ++c)
            bf[c] = *(const v2f*)(bptr + (size_t)c * 16 * FDIM + kk);   // global_load_b64
#pragma unroll
        for (int r = 0; r < 4; ++r)
#pragma unroll
            for (int c = 0; c < 4; ++c)
                // 8 args: (neg_a, A, neg_b, B, c_mod, C, reuse_a, reuse_b)
                acc[r][c] = __builtin_amdgcn_wmma_f32_16x16x4_f32(
                    false, af[r], false, bf[c], (short)0, acc[r][c], false, false);
    }

    // C/D layout: VGPR i -> row (lane>>4)*8 + i, col lane&15.
    const int mOff = (lane >> 4) * 8;
    const int col  = lane & 15;
    float* __restrict__ op =
        dst + ((size_t)(b * NHEADS + h) * S_PTS + sBase + mOff) * HD + col;

#pragma unroll
    for (int r = 0; r < 4; ++r)
#pragma unroll
        for (int i = 0; i < 8; ++i)
#pragma unroll
            for (int c = 0; c < 4; ++c)
                op[((size_t)r * 16 + i) * HD + c * 16] = acc[r][c][i];
}

// ---------------------------------------------------------------------------
// Kernel 2: neighbor attention. Block = 256 threads = 8 waves = one (b, s);
// wave h handles head h. Lane j owns neighbor j for the score phase;
// lane d owns output dims {d, d+32} for the weighted-V phase.
// q/k/v are in [B, H, S, Hd] fp32 (from kernel 1).
// ---------------------------------------------------------------------------
__global__ __launch_bounds__(256)
void nbr_attention(const float* __restrict__ qb,
                   const float* __restrict__ kb,
                   const float* __restrict__ vb,
                   const int*   __restrict__ nbr,
                   float*       __restrict__ out)
{
    __shared__ float qs[NHEADS * HD];       // 2 KB: q for all heads of (b,s)

    const int s = blockIdx.x;
    const int b = blockIdx.y;
    const int tid  = threadIdx.x;
    const int h    = tid >> 5;
    const int lane = tid & 31;

    // Stage q[b, :, s, :] into LDS (heads are strided by S*Hd in global).
#pragma unroll
    for (int c = 0; c < 2; ++c) {
        int idx = tid + c * 256;            // 0..511
        int hh  = idx >> 6;
        int dd  = idx & 63;
        qs[idx] = qb[((size_t)(b * NHEADS + hh) * S_PTS + s) * HD + dd];
    }
    __syncthreads();

    // ---- scores: lane j computes q . k[nbr_j] over 64 dims ----
    const int nj = nbr[(size_t)s * KNBR + lane];
    const float* __restrict__ kr =
        kb + ((size_t)(b * NHEADS + h) * S_PTS + nj) * HD;
    const float* __restrict__ qh = &qs[h * HD];

    float acc = 0.0f;
#pragma unroll
    for (int d = 0; d < HD; d += 4) {
        float4 kv = *(const float4*)(kr + d);   // lane-private contiguous 16B
        acc += qh[d + 0] * kv.x;                // LDS same-address broadcast
        acc += qh[d + 1] * kv.y;
        acc += qh[d + 2] * kv.z;
        acc += qh[d + 3] * kv.w;
    }
    float score = acc * 0.125f;                  // Hd^-0.5 = 1/8 exactly

    // ---- wave32 softmax over the 32 neighbors ----
    float mx = score;
#pragma unroll
    for (int off = 16; off > 0; off >>= 1)
        mx = fmaxf(mx, __shfl_xor(mx, off, 32));
    float e = __expf(score - mx);
    float sum = e;
#pragma unroll
    for (int off = 16; off > 0; off >>= 1)
        sum += __shfl_xor(sum, off, 32);
    const float w = e / sum;

    // ---- out[d] = sum_j w_j * v[nbr_j][d]; lane owns dims lane, lane+32 ----
    float o0 = 0.0f, o1 = 0.0f;
    const float* __restrict__ vhead = vb + ((size_t)(b * NHEADS + h) * S_PTS) * HD;
#pragma unroll 4
    for (int j = 0; j < KNBR; ++j) {
        float wj  = __shfl(w, j, 32);
        int   njj = __shfl(nj, j, 32);
        const float* __restrict__ vr = vhead + (size_t)njj * HD;
        o0 += wj * vr[lane];                     // coalesced 128B
        o1 += wj * vr[lane + 32];                // coalesced 128B
    }

    float* __restrict__ op = out + ((size_t)b * S_PTS + s) * FDIM + h * HD;
    op[lane]      = o0;
    op[lane + 32] = o1;
}

// ---------------------------------------------------------------------------
extern "C" void kernel_launch(void* const* d_in, const int* in_sizes, int n_in,
                              void* d_out, int out_size, void* d_ws, size_t ws_size,
                              hipStream_t stream)
{
    const float* x   = (const float*)d_in[0];
    const float* Wq  = (const float*)d_in[1];
    const float* Wk  = (const float*)d_in[2];
    const float* Wv  = (const float*)d_in[3];
    const int*   nbr = (const int*)d_in[4];
    float* out = (float*)d_out;

    const size_t per = (size_t)BATCH * NHEADS * S_PTS * HD;  // 16.384M floats
    float* qb = (float*)d_ws;
    float* kb = qb + per;
    float* vb = kb + per;

    dim3 g1(S_PTS / 64, NHEADS, BATCH * 3);      // 125 x 8 x 12 wave-tiles
    qkv_proj_wmma<<<g1, dim3(32), 0, stream>>>(x, Wq, Wk, Wv, qb, kb, vb);

    dim3 g2(S_PTS, BATCH);
    nbr_attention<<<g2, dim3(256), 0, stream>>>(qb, kb, vb, nbr, out);
}